// DGCNN_AE_36601711296670
// MI455X (gfx1250) — compile-verified
//
#include <hip/hip_runtime.h>

#define BN 8
#define NPTS 4096
#define KNN 20
#define NEG_SLOPE 0.2f
#define EPSV 1e-5f
#define NEG_INF (-3.0e38f)

typedef float v2f __attribute__((ext_vector_type(2)));
typedef float v8f __attribute__((ext_vector_type(8)));

__device__ __forceinline__ v8f wmma4(v2f a, v2f b, v8f c) {
  // D = A(16x4) * B(4x16) + C, fp32
  return __builtin_amdgcn_wmma_f32_16x16x4_f32(false, a, false, b, (short)0, c, false, false);
}

__device__ __forceinline__ float lrelu(float y) { return y >= 0.f ? y : NEG_SLOPE * y; }

__global__ void zero_kernel(float* p, int n) {
  int i = blockIdx.x * blockDim.x + threadIdx.x;
  if (i < n) p[i] = 0.f;
}

template <int C>
__global__ void norms_kernel(const float* __restrict__ x, float* __restrict__ nrm) {
  int t = blockIdx.x * blockDim.x + threadIdx.x;
  if (t >= BN * NPTS) return;
  const float* p = x + (size_t)t * C;
  float s = 0.f;
#pragma unroll
  for (int c = 0; c < C; ++c) s += p[c] * p[c];
  nrm[t] = s;
}

// ---------------- kNN: WMMA distance tiles + iterative wave argmax ----------------
template <int CIN, int KA>  // KA = K padded to multiple of 4 (>= CIN)
__global__ void __launch_bounds__(256) knn_kernel(const float* __restrict__ xin,
                                                  const float* __restrict__ nrm,
                                                  int* __restrict__ knn_out) {
  __shared__ float rowf[16][KA + 1];
  __shared__ float rowxx[16];
  __shared__ float dist[16][1024];
  __shared__ float runv[16][KNN];
  __shared__ int   runi[16][KNN];
  __shared__ float cndv[16][KNN];
  __shared__ int   cndi[16][KNN];
  __shared__ float mrgv[16][KNN];
  __shared__ int   mrgi[16][KNN];

  const int b = blockIdx.y;
  const int r0 = blockIdx.x * 16;
  const int tid = threadIdx.x;
  const int lane = tid & 31, wave = tid >> 5;
  const int half = lane >> 4, lm = lane & 15;

  for (int i = tid; i < 16 * (KA + 1); i += 256) ((float*)rowf)[i] = 0.f;
  __syncthreads();
  for (int i = tid; i < 16 * CIN; i += 256) {
    int r = i / CIN, c = i % CIN;
    rowf[r][c] = xin[((size_t)(b * NPTS + r0 + r)) * CIN + c];
  }
  if (tid < 16) rowxx[tid] = nrm[b * NPTS + r0 + tid];
  for (int i = tid; i < 16 * KNN; i += 256) {
    ((float*)runv)[i] = NEG_INF;
    ((int*)runi)[i] = 0x7fffffff;
  }
  __syncthreads();

  for (int chunk = 0; chunk < NPTS / 1024; ++chunk) {
    const int cb = chunk * 1024;
    // --- distance tiles: 64 col-tiles of 16, 8 per wave ---
    for (int t8 = 0; t8 < 8; ++t8) {
      const int ct = wave + t8 * 8;
      const int col = cb + ct * 16 + lm;
      const float xxc = nrm[b * NPTS + col];
      const float* colp = xin + (size_t)(b * NPTS + col) * CIN;
      v8f acc = {0.f, 0.f, 0.f, 0.f, 0.f, 0.f, 0.f, 0.f};
#pragma unroll
      for (int kk = 0; kk < KA / 4; ++kk) {
        const int cbase = kk * 4 + 2 * half;
        v2f a, bb;
        a[0] = rowf[lm][cbase + 0];
        a[1] = rowf[lm][cbase + 1];
        if (CIN % 4 == 0) {  // compile-time: aligned float2 global load
          float2 cv = *(const float2*)(colp + cbase);
          bb[0] = cv.x; bb[1] = cv.y;
        } else {
          bb[0] = (cbase + 0 < CIN) ? colp[cbase + 0] : 0.f;
          bb[1] = (cbase + 1 < CIN) ? colp[cbase + 1] : 0.f;
        }
        acc = wmma4(a, bb, acc);
      }
#pragma unroll
      for (int v = 0; v < 8; ++v) {
        int m = v + 8 * half;
        dist[m][ct * 16 + lm] = 2.f * acc[v] - rowxx[m] - xxc;
      }
    }
    __syncthreads();
    // --- per-row top-20 within chunk, then merge into running list ---
    for (int rr = 0; rr < 2; ++rr) {
      const int row = wave * 2 + rr;
      volatile float* drow = &dist[row][0];
      for (int it = 0; it < KNN; ++it) {
        float bv = NEG_INF;
        int bi = 0x7fffffff;
        for (int j = lane; j < 1024; j += 32) {
          float v = drow[j];
          int ci = cb + j;
          if (v > bv || (v == bv && ci < bi)) { bv = v; bi = ci; }
        }
#pragma unroll
        for (int off = 16; off > 0; off >>= 1) {
          float ov = __shfl_xor(bv, off, 32);
          int oi = __shfl_xor(bi, off, 32);
          if (ov > bv || (ov == bv && oi < bi)) { bv = ov; bi = oi; }
        }
        if (lane == 0) {
          cndv[row][it] = bv;
          cndi[row][it] = bi;
          drow[bi - cb] = NEG_INF;  // mask winner (volatile, DS in-order per wave)
        }
      }
      if (lane == 0) {  // stable merge of two sorted-desc lists, keep 20
        int pi = 0, qi = 0;
        for (int t = 0; t < KNN; ++t) {
          float pv = runv[row][pi], qv = cndv[row][qi];
          bool takeP = (pv > qv) || (pv == qv && runi[row][pi] <= cndi[row][qi]);
          if (takeP) { mrgv[row][t] = pv; mrgi[row][t] = runi[row][pi]; ++pi; }
          else       { mrgv[row][t] = qv; mrgi[row][t] = cndi[row][qi]; ++qi; }
        }
        for (int t = 0; t < KNN; ++t) { runv[row][t] = mrgv[row][t]; runi[row][t] = mrgi[row][t]; }
      }
    }
    __syncthreads();
  }
  for (int rr = 0; rr < 2; ++rr) {
    const int row = wave * 2 + rr;
    if (lane < KNN)
      knn_out[((size_t)(b * NPTS + r0 + row)) * KNN + lane] = runi[row][lane];
  }
}

// ------------- Fused edge-conv: gather + WMMA conv (+conv) + BN/LReLU + max over k -------------
template <int CIN, int KA, bool DOUBLE>  // KA = 2*CIN padded to mult of 4
__global__ void __launch_bounds__(256) edgeconv_kernel(
    const float* __restrict__ xin, const int* __restrict__ knn_idx,
    const float* __restrict__ Wa, const float* __restrict__ bna,
    const float* __restrict__ Wb, const float* __restrict__ bnb,
    float* __restrict__ xout) {
  __shared__ float WaS[64][KA + 1];
  __shared__ float WbS[DOUBLE ? 64 : 1][65];
  __shared__ float cenS[16][CIN + 1];
  __shared__ float nbrS[64][CIN + 1];
  __shared__ float h1S[DOUBLE ? 64 : 1][65];
  __shared__ float maxb[16][64];
  __shared__ float sA[64], tA[64], sB[64], tB[64];

  const int b = blockIdx.y;
  const int p0 = blockIdx.x * 16;
  const int tid = threadIdx.x, lane = tid & 31, wave = tid >> 5;
  const int half = lane >> 4, lm = lane & 15;

  for (int i = tid; i < 64 * (KA + 1); i += 256) ((float*)WaS)[i] = 0.f;
  __syncthreads();
  for (int i = tid; i < 64 * 2 * CIN; i += 256) WaS[i / (2 * CIN)][i % (2 * CIN)] = Wa[i];
  if (DOUBLE)
    for (int i = tid; i < 64 * 64; i += 256) WbS[i >> 6][i & 63] = Wb[i];
  if (tid < 64) {
    float g = bna[tid], be = bna[64 + tid], mn = bna[128 + tid], vr = bna[192 + tid];
    float s = g * rsqrtf(vr + EPSV);
    sA[tid] = s; tA[tid] = be - mn * s;
    if (DOUBLE) {
      g = bnb[tid]; be = bnb[64 + tid]; mn = bnb[128 + tid]; vr = bnb[192 + tid];
      s = g * rsqrtf(vr + EPSV);
      sB[tid] = s; tB[tid] = be - mn * s;
    }
  }
  for (int i = tid; i < 16 * CIN; i += 256)
    cenS[i / CIN][i % CIN] = xin[((size_t)(b * NPTS + p0 + i / CIN)) * CIN + i % CIN];
  for (int i = tid; i < 16 * 64; i += 256) ((float*)maxb)[i] = NEG_INF;
  __syncthreads();

  for (int grp = 0; grp < KNN / 4; ++grp) {
    for (int i = tid; i < 64 * CIN; i += 256) {
      int e = i / CIN, c = i % CIN;
      int nb = knn_idx[((size_t)(b * NPTS + p0 + (e >> 2))) * KNN + grp * 4 + (e & 3)];
      nbrS[e][c] = xin[((size_t)(b * NPTS + nb)) * CIN + c];
    }
    __syncthreads();
    // conv A over 64 edges: 4 M-tiles x 4 N-tiles, 2 jobs/wave
    for (int jj = 0; jj < 2; ++jj) {
      int job = wave + jj * 8;
      int mt = job & 3, nt = job >> 2;
      int e = mt * 16 + lm, pp = e >> 2;
      v8f acc = {0.f, 0.f, 0.f, 0.f, 0.f, 0.f, 0.f, 0.f};
      if (CIN % 4 == 0) {
        // phase 1: k in [0,CIN): f = nbr - cen  (no per-element selects)
#pragma unroll 4
        for (int kk = 0; kk < CIN / 4; ++kk) {
          int cb2 = kk * 4 + 2 * half;
          v2f a, bb;
          a[0] = nbrS[e][cb2 + 0] - cenS[pp][cb2 + 0];
          a[1] = nbrS[e][cb2 + 1] - cenS[pp][cb2 + 1];
          bb[0] = WaS[nt * 16 + lm][cb2 + 0];
          bb[1] = WaS[nt * 16 + lm][cb2 + 1];
          acc = wmma4(a, bb, acc);
        }
        // phase 2: k in [CIN,2CIN): f = cen
#pragma unroll 4
        for (int kk = 0; kk < CIN / 4; ++kk) {
          int cb2 = kk * 4 + 2 * half;
          v2f a, bb;
          a[0] = cenS[pp][cb2 + 0];
          a[1] = cenS[pp][cb2 + 1];
          bb[0] = WaS[nt * 16 + lm][CIN + cb2 + 0];
          bb[1] = WaS[nt * 16 + lm][CIN + cb2 + 1];
          acc = wmma4(a, bb, acc);
        }
      } else {  // generic masked path (stage 1, CIN=3, KA=8)
#pragma unroll
        for (int kk = 0; kk < KA / 4; ++kk) {
          int cb2 = kk * 4 + 2 * half;
          v2f a, bb;
#pragma unroll
          for (int r = 0; r < 2; ++r) {
            int c = cb2 + r;
            float f;
            if (c < CIN) f = nbrS[e][c] - cenS[pp][c];
            else if (c < 2 * CIN) f = cenS[pp][c - CIN];
            else f = 0.f;
            a[r] = f;
            bb[r] = WaS[nt * 16 + lm][c];
          }
          acc = wmma4(a, bb, acc);
        }
      }
      int n = nt * 16 + lm;
      float s = sA[n], t = tA[n];
      if (DOUBLE) {
#pragma unroll
        for (int v = 0; v < 8; ++v)
          h1S[mt * 16 + v + 8 * half][n] = lrelu(acc[v] * s + t);
      } else {
        float c1 = NEG_INF, c2 = NEG_INF;
#pragma unroll
        for (int v = 0; v < 4; ++v) c1 = fmaxf(c1, lrelu(acc[v] * s + t));
#pragma unroll
        for (int v = 4; v < 8; ++v) c2 = fmaxf(c2, lrelu(acc[v] * s + t));
        int pA = mt * 4 + 2 * half;  // owner-unique (p,n) across all lanes/jobs
        maxb[pA][n] = fmaxf(maxb[pA][n], c1);
        maxb[pA + 1][n] = fmaxf(maxb[pA + 1][n], c2);
      }
    }
    __syncthreads();
    if (DOUBLE) {
      for (int jj = 0; jj < 2; ++jj) {
        int job = wave + jj * 8;
        int mt = job & 3, nt = job >> 2;
        v8f acc = {0.f, 0.f, 0.f, 0.f, 0.f, 0.f, 0.f, 0.f};
#pragma unroll 4
        for (int kk = 0; kk < 16; ++kk) {
          int cb2 = kk * 4 + 2 * half;
          v2f a, bb;
          a[0] = h1S[mt * 16 + lm][cb2 + 0];
          a[1] = h1S[mt * 16 + lm][cb2 + 1];
          bb[0] = WbS[nt * 16 + lm][cb2 + 0];
          bb[1] = WbS[nt * 16 + lm][cb2 + 1];
          acc = wmma4(a, bb, acc);
        }
        int n = nt * 16 + lm;
        float s = sB[n], t = tB[n];
        float c1 = NEG_INF, c2 = NEG_INF;
#pragma unroll
        for (int v = 0; v < 4; ++v) c1 = fmaxf(c1, lrelu(acc[v] * s + t));
#pragma unroll
        for (int v = 4; v < 8; ++v) c2 = fmaxf(c2, lrelu(acc[v] * s + t));
        int pA = mt * 4 + 2 * half;
        maxb[pA][n] = fmaxf(maxb[pA][n], c1);
        maxb[pA + 1][n] = fmaxf(maxb[pA + 1][n], c2);
      }
    }
    __syncthreads();
  }
  for (int i = tid; i < 16 * 64; i += 256)
    xout[((size_t)(b * NPTS + p0 + (i >> 6))) * 64 + (i & 63)] = maxb[i >> 6][i & 63];
}

// ------------- conv6 (1024x192) + mean accumulation over N -------------
__global__ void __launch_bounds__(256) conv6_kernel(
    const float* __restrict__ x1, const float* __restrict__ x2, const float* __restrict__ x3,
    const float* __restrict__ W6, const float* __restrict__ bn6,
    float* __restrict__ gacc) {
  __shared__ float catf[16][193];
  __shared__ float s6[1024], t6[1024];
  const int b = blockIdx.y, p0 = blockIdx.x * 16;
  const int tid = threadIdx.x, lane = tid & 31, wave = tid >> 5;
  const int half = lane >> 4, lm = lane & 15;

  for (int i = tid; i < 1024; i += 256) {
    float g = bn6[i], be = bn6[1024 + i], mn = bn6[2048 + i], vr = bn6[3072 + i];
    float s = g * rsqrtf(vr + EPSV);
    s6[i] = s; t6[i] = be - mn * s;
  }
  for (int i = tid; i < 16 * 192; i += 256) {
    int p = i / 192, c = i % 192;
    const float* src = c < 64 ? x1 : (c < 128 ? x2 : x3);
    catf[p][c] = src[((size_t)(b * NPTS + p0 + p)) * 64 + (c & 63)];
  }
  __syncthreads();
  for (int jj = 0; jj < 8; ++jj) {
    int n = (wave + jj * 8) * 16 + lm;
    const float* wrow = W6 + (size_t)n * 192;
    v8f acc = {0.f, 0.f, 0.f, 0.f, 0.f, 0.f, 0.f, 0.f};
#pragma unroll 4
    for (int kk = 0; kk < 48; ++kk) {
      int cb2 = kk * 4 + 2 * half;
      v2f a, bb;
      a[0] = catf[lm][cb2];
      a[1] = catf[lm][cb2 + 1];
      float2 wv = *(const float2*)(wrow + cb2);
      bb[0] = wv.x; bb[1] = wv.y;
      acc = wmma4(a, bb, acc);
    }
    float s = s6[n], t = t6[n], sum = 0.f;
#pragma unroll
    for (int v = 0; v < 8; ++v) sum += lrelu(acc[v] * s + t);
    sum += __shfl_xor(sum, 16, 32);  // combine the two 8-row halves
    if (half == 0) atomicAdd(&gacc[b * 1024 + n], sum);
  }
}

__global__ void finalize_g(const float* __restrict__ gacc, float* __restrict__ gbuf,
                           float* __restrict__ dout) {
  int i = blockIdx.x * blockDim.x + threadIdx.x;
  if (i < BN * 1024) {
    float v = gacc[i] * (1.f / NPTS);
    gbuf[i] = v;
    dout[i] = v;
  }
}

// ------------- head: W7(512x1216) -> W8(256x512) -> W9(3x256) -------------
__global__ void __launch_bounds__(256) head_kernel(
    const float* __restrict__ gbuf,
    const float* __restrict__ x1, const float* __restrict__ x2, const float* __restrict__ x3,
    const float* __restrict__ W7, const float* __restrict__ bn7,
    const float* __restrict__ W8, const float* __restrict__ bn8,
    const float* __restrict__ W9, float* __restrict__ outp) {
  __shared__ float gl[1024];
  __shared__ float xt[3][16][65];
  __shared__ float h7[16][513];
  __shared__ float h8[16][257];
  __shared__ float s7[512], t7[512], s8[256], t8v[256];

  const int b = blockIdx.y, p0 = blockIdx.x * 16;
  const int tid = threadIdx.x, lane = tid & 31, wave = tid >> 5;
  const int half = lane >> 4, lm = lane & 15;

  for (int i = tid; i < 1024; i += 256) gl[i] = gbuf[b * 1024 + i];
  for (int i = tid; i < 512; i += 256) {
    float g = bn7[i], be = bn7[512 + i], mn = bn7[1024 + i], vr = bn7[1536 + i];
    float s = g * rsqrtf(vr + EPSV);
    s7[i] = s; t7[i] = be - mn * s;
  }
  {
    int i = tid;
    float g = bn8[i], be = bn8[256 + i], mn = bn8[512 + i], vr = bn8[768 + i];
    float s = g * rsqrtf(vr + EPSV);
    s8[i] = s; t8v[i] = be - mn * s;
  }
  for (int i = tid; i < 3 * 16 * 64; i += 256) {
    int w = i / (16 * 64), rem = i % (16 * 64), p = rem / 64, c = rem % 64;
    const float* src = w == 0 ? x1 : (w == 1 ? x2 : x3);
    xt[w][p][c] = src[((size_t)(b * NPTS + p0 + p)) * 64 + c];
  }
  __syncthreads();

  // conv7: K = 1216 = [g (1024) | x1 | x2 | x3 (192)], split at the boundary
  for (int jj = 0; jj < 4; ++jj) {
    int n = (wave + jj * 8) * 16 + lm;
    const float* wrow = W7 + (size_t)n * 1216;
    v8f acc = {0.f, 0.f, 0.f, 0.f, 0.f, 0.f, 0.f, 0.f};
    for (int kk = 0; kk < 256; ++kk) {  // k in [0,1024): A = g (broadcast)
      int cb2 = kk * 4 + 2 * half;
      v2f a, bb;
      a[0] = gl[cb2];
      a[1] = gl[cb2 + 1];
      float2 wv = *(const float2*)(wrow + cb2);
      bb[0] = wv.x; bb[1] = wv.y;
      acc = wmma4(a, bb, acc);
    }
#pragma unroll 4
    for (int kk = 0; kk < 48; ++kk) {  // k in [1024,1216): A = xt
      int c2 = kk * 4 + 2 * half;      // 0..190, even; never crosses 64-boundary within pair
      int w = c2 >> 6, cc = c2 & 63;
      v2f a, bb;
      a[0] = xt[w][lm][cc + 0];
      a[1] = xt[w][lm][cc + 1];
      float2 wv = *(const float2*)(wrow + 1024 + c2);
      bb[0] = wv.x; bb[1] = wv.y;
      acc = wmma4(a, bb, acc);
    }
    float s = s7[n], t = t7[n];
#pragma unroll
    for (int v = 0; v < 8; ++v) h7[v + 8 * half][n] = lrelu(acc[v] * s + t);
  }
  __syncthreads();
  // conv8: K = 512
  for (int jj = 0; jj < 2; ++jj) {
    int n = (wave + jj * 8) * 16 + lm;
    const float* wrow = W8 + (size_t)n * 512;
    v8f acc = {0.f, 0.f, 0.f, 0.f, 0.f, 0.f, 0.f, 0.f};
#pragma unroll 4
    for (int kk = 0; kk < 128; ++kk) {
      int cb2 = kk * 4 + 2 * half;
      v2f a, bb;
      a[0] = h7[lm][cb2];
      a[1] = h7[lm][cb2 + 1];
      float2 wv = *(const float2*)(wrow + cb2);
      bb[0] = wv.x; bb[1] = wv.y;
      acc = wmma4(a, bb, acc);
    }
    float s = s8[n], t = t8v[n];
#pragma unroll
    for (int v = 0; v < 8; ++v) h8[v + 8 * half][n] = lrelu(acc[v] * s + t);
  }
  __syncthreads();
  // W9: 16 points x 3 outs, plain dot products (N=3 too thin for WMMA)
  if (tid < 48) {
    int p = tid / 3, o = tid % 3;
    const float* wr = W9 + o * 256;
    float s = 0.f;
    for (int c = 0; c < 256; ++c) s += h8[p][c] * wr[c];
    outp[((size_t)(b * NPTS + p0 + p)) * 3 + o] = s;
  }
}

extern "C" void kernel_launch(void* const* d_in, const int* in_sizes, int n_in,
                              void* d_out, int out_size, void* d_ws, size_t ws_size,
                              hipStream_t stream) {
  (void)in_sizes; (void)n_in; (void)out_size; (void)ws_size;
  const float* x  = (const float*)d_in[0];
  const float* W1 = (const float*)d_in[1];
  const float* W2 = (const float*)d_in[2];
  const float* W3 = (const float*)d_in[3];
  const float* W4 = (const float*)d_in[4];
  const float* W5 = (const float*)d_in[5];
  const float* W6 = (const float*)d_in[6];
  const float* W7 = (const float*)d_in[7];
  const float* W8 = (const float*)d_in[8];
  const float* W9 = (const float*)d_in[9];
  const float* bn1 = (const float*)d_in[10];
  const float* bn2 = (const float*)d_in[11];
  const float* bn3 = (const float*)d_in[12];
  const float* bn4 = (const float*)d_in[13];
  const float* bn5 = (const float*)d_in[14];
  const float* bn6 = (const float*)d_in[15];
  const float* bn7 = (const float*)d_in[16];
  const float* bn8 = (const float*)d_in[17];

  // workspace partition (~33 MB)
  float* nrm  = (float*)d_ws;                       // B*N
  int*   idx1 = (int*)(nrm + BN * NPTS);
  int*   idx2 = idx1 + (size_t)BN * NPTS * KNN;
  int*   idx3 = idx2 + (size_t)BN * NPTS * KNN;
  float* x1   = (float*)(idx3 + (size_t)BN * NPTS * KNN);
  float* x2   = x1 + (size_t)BN * NPTS * 64;
  float* x3   = x2 + (size_t)BN * NPTS * 64;
  float* gacc = x3 + (size_t)BN * NPTS * 64;
  float* gbuf = gacc + BN * 1024;

  dim3 blk(256);
  dim3 gridN(NPTS / 16, BN);
  int nBN = BN * NPTS;

  zero_kernel<<<(BN * 1024 + 255) / 256, blk, 0, stream>>>(gacc, BN * 1024);

  norms_kernel<3><<<(nBN + 255) / 256, blk, 0, stream>>>(x, nrm);
  knn_kernel<3, 4><<<gridN, blk, 0, stream>>>(x, nrm, idx1);
  edgeconv_kernel<3, 8, true><<<gridN, blk, 0, stream>>>(x, idx1, W1, bn1, W2, bn2, x1);

  norms_kernel<64><<<(nBN + 255) / 256, blk, 0, stream>>>(x1, nrm);
  knn_kernel<64, 64><<<gridN, blk, 0, stream>>>(x1, nrm, idx2);
  edgeconv_kernel<64, 128, true><<<gridN, blk, 0, stream>>>(x1, idx2, W3, bn3, W4, bn4, x2);

  norms_kernel<64><<<(nBN + 255) / 256, blk, 0, stream>>>(x2, nrm);
  knn_kernel<64, 64><<<gridN, blk, 0, stream>>>(x2, nrm, idx3);
  edgeconv_kernel<64, 128, false><<<gridN, blk, 0, stream>>>(x2, idx3, W5, bn5, nullptr, nullptr, x3);

  conv6_kernel<<<gridN, blk, 0, stream>>>(x1, x2, x3, W6, bn6, gacc);
  finalize_g<<<(BN * 1024 + 255) / 256, blk, 0, stream>>>(gacc, gbuf, (float*)d_out);
  head_kernel<<<gridN, blk, 0, stream>>>(gbuf, x1, x2, x3, W7, bn7, W8, bn8, W9,
                                         (float*)d_out + BN * 1024);
}